// ChebNet_69604239999350
// MI455X (gfx1250) — compile-verified
//
#include <hip/hip_runtime.h>
#include <math.h>

#define NNODES 50000
#define NEDGES 800000
#define IN_F   128
#define HID_F  128
#define OUT_F  64

typedef float v2f __attribute__((ext_vector_type(2)));
typedef float v8f __attribute__((ext_vector_type(8)));

// ---------------- utility kernels ----------------

__global__ void zero_f4(float4* __restrict__ p, int n4) {
    int i = blockIdx.x * blockDim.x + threadIdx.x;
    if (i < n4) p[i] = make_float4(0.f, 0.f, 0.f, 0.f);
}

__global__ void deg_kernel(const int* __restrict__ dst, float* __restrict__ deg, int ne) {
    int e = blockIdx.x * blockDim.x + threadIdx.x;
    if (e < ne) atomicAdd(&deg[dst[e]], 1.0f);
}

// in place: deg -> 1/sqrt(max(deg,1))
__global__ void dinv_kernel(float* __restrict__ deg, int n) {
    int i = blockIdx.x * blockDim.x + threadIdx.x;
    if (i < n) deg[i] = 1.0f / sqrtf(fmaxf(deg[i], 1.0f));
}

// one wave per edge; lane handles 4 consecutive features (32*4 = 128)
// h[dst] += x[src] * dinv[src]
__global__ void edge_scatter(const float* __restrict__ x,
                             const int* __restrict__ src,
                             const int* __restrict__ dst,
                             const float* __restrict__ dinv,
                             float* __restrict__ h) {
    int e = (blockIdx.x * blockDim.x + threadIdx.x) >> 5;
    int lane = threadIdx.x & 31;
    int s = src[e];
    int d = dst[e];
    float sc = dinv[s];
    const float4 v = *(const float4*)(x + (size_t)s * 128 + lane * 4);
    float* hp = h + (size_t)d * 128 + lane * 4;
    atomicAdd(hp + 0, v.x * sc);
    atomicAdd(hp + 1, v.y * sc);
    atomicAdd(hp + 2, v.z * sc);
    atomicAdd(hp + 3, v.w * sc);
}

// x1 = -h * dinv[row]   (re_norm == 1 so the x*(re_norm-1) term vanishes)
__global__ void neg_scale(float* __restrict__ h, const float* __restrict__ dinv) {
    int t = blockIdx.x * blockDim.x + threadIdx.x;   // N*32 threads
    int n = t >> 5;
    int c = (t & 31) << 2;
    float sc = -dinv[n];
    float4* p = (float4*)(h + (size_t)n * 128 + c);
    float4 v = *p;
    v.x *= sc; v.y *= sc; v.z *= sc; v.w *= sc;
    *p = v;
}

// ---------------- WMMA f32 GEMM ----------------
// out[N, NC] = concat(A1, A2)[N, 256] @ W[256, NC] + bias
// block = 32 * (NC/16) threads, one wave per 16-col tile; grid = N/16 blocks.
template <int NC>
__global__ __launch_bounds__(32 * (NC / 16)) void cheb_gemm(
        const float* __restrict__ A1, const float* __restrict__ A2,
        const float* __restrict__ W,  const float* __restrict__ bias,
        float* __restrict__ out) {
    __shared__ float As[16 * 260];             // 16 rows x 256, padded stride 260

    const int mbase = blockIdx.x * 16;
    const int tid = threadIdx.x;

    // cooperatively stage 16x256 A tile (first 128 cols from A1, rest from A2)
    for (int i = tid; i < 16 * 64; i += blockDim.x) {   // 1024 float4's
        int row = i >> 6;
        int col = (i & 63) << 2;
        const float* sp = (col < 128)
                        ? (A1 + (size_t)(mbase + row) * 128 + col)
                        : (A2 + (size_t)(mbase + row) * 128 + (col - 128));
        float4 v = *(const float4*)sp;
        *(float4*)(&As[row * 260 + col]) = v;
    }
    __syncthreads();

    const int ntile = tid >> 5;                // wave id == N-tile
    const int lane  = tid & 31;
    const int half  = lane >> 4;               // 0: K+{0,1} | rows r ; 1: K+{2,3} | rows r+8
    const int l     = lane & 15;

    v8f c = {};
    #pragma unroll 4
    for (int k0 = 0; k0 < 256; k0 += 4) {
        // A frag (16x4 f32): lane l holds row M=l, K = k0 + 2*half + {0,1}
        v2f a = *(const v2f*)(&As[l * 260 + k0 + 2 * half]);
        // B frag (4x16 f32): lane l holds col N=l, K = k0 + 2*half + {0,1}
        const float* wp = W + (size_t)(k0 + 2 * half) * NC + ntile * 16 + l;
        v2f b;
        b.x = wp[0];
        b.y = wp[NC];
        c = __builtin_amdgcn_wmma_f32_16x16x4_f32(
                false, a, false, b, (short)0, c, false, false);
    }

    const int col = ntile * 16 + l;
    const float bv = bias[col];
    #pragma unroll
    for (int r = 0; r < 8; ++r) {
        int row = mbase + r + 8 * half;        // C/D layout: VGPR r -> rows r / r+8
        out[(size_t)row * NC + col] = c[r] + bv;
    }
}

// ---------------- launch ----------------

extern "C" void kernel_launch(void* const* d_in, const int* in_sizes, int n_in,
                              void* d_out, int out_size, void* d_ws, size_t ws_size,
                              hipStream_t stream) {
    const float* feat = (const float*)d_in[0];   // [N,128]
    const int*   src  = (const int*)d_in[1];     // [E]
    const int*   dst  = (const int*)d_in[2];     // [E]
    const float* W1   = (const float*)d_in[3];   // [256,128]
    const float* b1   = (const float*)d_in[4];   // [128]
    const float* W2   = (const float*)d_in[5];   // [256,64]
    const float* b2   = (const float*)d_in[6];   // [64]
    float* out = (float*)d_out;                  // [N,64]

    char* ws = (char*)d_ws;
    float* dinv = (float*)ws;                                  // N floats
    float* h    = (float*)(ws + 200704);                       // N*128 floats
    float* out1 = (float*)(ws + 200704 + (size_t)NNODES * 128 * 4); // N*128 floats

    const int TB = 256;

    // degree -> dinv (in place)
    zero_f4<<<(NNODES / 4 + TB - 1) / TB, TB, 0, stream>>>((float4*)dinv, NNODES / 4);
    deg_kernel<<<(NEDGES + TB - 1) / TB, TB, 0, stream>>>(dst, dinv, NEDGES);
    dinv_kernel<<<(NNODES + TB - 1) / TB, TB, 0, stream>>>(dinv, NNODES);

    const int n4h   = NNODES * 128 / 4;           // float4 count of h
    const int scat  = NEDGES * 32;                // threads for edge_scatter
    const int nsc   = NNODES * 32;                // threads for neg_scale

    // ---- layer 1 ----
    zero_f4<<<(n4h + TB - 1) / TB, TB, 0, stream>>>((float4*)h, n4h);
    edge_scatter<<<scat / TB, TB, 0, stream>>>(feat, src, dst, dinv, h);
    neg_scale<<<nsc / TB, TB, 0, stream>>>(h, dinv);
    cheb_gemm<HID_F><<<NNODES / 16, 32 * (HID_F / 16), 0, stream>>>(feat, h, W1, b1, out1);

    // ---- layer 2 ----
    zero_f4<<<(n4h + TB - 1) / TB, TB, 0, stream>>>((float4*)h, n4h);
    edge_scatter<<<scat / TB, TB, 0, stream>>>(out1, src, dst, dinv, h);
    neg_scale<<<nsc / TB, TB, 0, stream>>>(h, dinv);
    cheb_gemm<OUT_F><<<NNODES / 16, 32 * (OUT_F / 16), 0, stream>>>(out1, h, W2, b2, out);
}